// GridMAX_11836929868047
// MI455X (gfx1250) — compile-verified
//
#include <hip/hip_runtime.h>

// out = max(cummax(x, axis=2), -1000.0f)  on x: [B=2, Tt=512, Ts=512, C=256] f32.
//
// Pure streaming op: 1 GiB compulsory traffic -> ~46 us floor at 23.3 TB/s.
// No matrix formulation exists for an order-statistic scan (max is not
// linear), so no WMMA. The win comes from b128 coalesced vector memory,
// non-temporal cache hints (each byte touched exactly once; don't pollute
// the 192 MB L2), and enough wave32s x outstanding loads to cover HBM
// latency (~16 MB in flight here).

typedef float v4f __attribute__((ext_vector_type(4)));

constexpr int kB  = 2;
constexpr int kTt = 512;
constexpr int kTs = 512;
constexpr int kC  = 256;
constexpr int kCV = kC / 4;          // 64 float4 slots across the C axis
constexpr int kRowsPerBlock = 4;     // 4 (b,tt) rows per block -> 256 thr = 8 wave32
constexpr float kPad = -1000.0f;

__global__ __launch_bounds__(kCV * kRowsPerBlock)
void cummax_ts_kernel(const v4f* __restrict__ x, v4f* __restrict__ out) {
    // threadIdx.x in [0,64): which float4 of the channel axis (contiguous in
    // memory -> lanes 0..31 of a wave cover 512 contiguous bytes).
    // threadIdx.y in [0,4): which (b,tt) row inside this block.
    const int c4  = threadIdx.x;
    const int row = blockIdx.x * kRowsPerBlock + threadIdx.y;   // b*Tt + tt

    const size_t base = (size_t)row * (size_t)kTs * (size_t)kCV + (size_t)c4;
    const v4f* __restrict__ px = x   + base;
    v4f*       __restrict__ po = out + base;

    // Running max seeded with the pad value implements the final clamp:
    // max(cummax(x), PAD) == scan starting from PAD.
    v4f m = { kPad, kPad, kPad, kPad };

    // Serial scan over Ts. The cross-iteration dependency is only the v_max
    // chain (registers); the 16 unrolled b128 loads are independent and can
    // all be outstanding (LOADcnt is 6-bit). Stores are fire-and-forget.
#pragma unroll 16
    for (int ts = 0; ts < kTs; ++ts) {
        v4f v = __builtin_nontemporal_load(px);     // TH=NT load, streamed once
        m.x = __builtin_fmaxf(m.x, v.x);
        m.y = __builtin_fmaxf(m.y, v.y);
        m.z = __builtin_fmaxf(m.z, v.z);
        m.w = __builtin_fmaxf(m.w, v.w);
        __builtin_nontemporal_store(m, po);         // TH=NT store, full lines
        px += kCV;                                  // stride C floats along Ts
        po += kCV;
    }
}

extern "C" void kernel_launch(void* const* d_in, const int* in_sizes, int n_in,
                              void* d_out, int out_size, void* d_ws, size_t ws_size,
                              hipStream_t stream) {
    (void)in_sizes; (void)n_in; (void)out_size; (void)d_ws; (void)ws_size;

    const v4f* x  = (const v4f*)d_in[0];
    v4f*      out = (v4f*)d_out;

    dim3 block(kCV, kRowsPerBlock, 1);                    // 64 x 4 = 256 threads
    dim3 grid((kB * kTt) / kRowsPerBlock, 1, 1);          // 256 blocks, 2048 waves
    hipLaunchKernelGGL(cummax_ts_kernel, grid, block, 0, stream, x, out);
}